// ModelNew_3556232922178
// MI455X (gfx1250) — compile-verified
//
#include <hip/hip_runtime.h>
#include <math.h>

typedef __attribute__((ext_vector_type(2))) float v2f;
typedef __attribute__((ext_vector_type(4))) float v4f;
typedef __attribute__((ext_vector_type(8))) float v8f;

// ---------------------------------------------------------------------------
// Phase 1: Sx[b][c] = sum over 32^3 spatial elements of x[b,c,:,:,:]
// One block per (b,c) slab (32768 contiguous floats = 8192 float4).
// Streaming, single-use data -> non-temporal b128 loads.
// ---------------------------------------------------------------------------
__global__ __launch_bounds__(256) void reduce_x_kernel(const float* __restrict__ x,
                                                       float* __restrict__ sx) {
    __shared__ float red[256];
    const int bc = blockIdx.x;  // b*32 + c, 0..511
    const v4f* __restrict__ base =
        reinterpret_cast<const v4f*>(x + (size_t)bc * 32768u);

    v4f acc = {};
#pragma unroll
    for (int i = 0; i < 32; ++i) {
        v4f v = __builtin_nontemporal_load(&base[threadIdx.x + i * 256]);
        acc += v;
    }
    red[threadIdx.x] = acc.x + acc.y + acc.z + acc.w;
    __syncthreads();
#pragma unroll
    for (int off = 128; off > 0; off >>= 1) {
        if ((int)threadIdx.x < off) red[threadIdx.x] += red[threadIdx.x + off];
        __syncthreads();
    }
    if (threadIdx.x == 0) sx[bc] = red[0];
}

// ---------------------------------------------------------------------------
// Phase 2: Sw[o][c] = sum of the 27 kernel taps of weight[o,c,:,:,:]
// (flip in the reference is irrelevant for a full-tap sum)
// ---------------------------------------------------------------------------
__global__ __launch_bounds__(256) void reduce_w_kernel(const float* __restrict__ w,
                                                       float* __restrict__ sw) {
    const int idx = blockIdx.x * 256 + threadIdx.x;  // 0..2047 = o*32 + c
    if (idx < 2048) {
        const float* __restrict__ p = w + (size_t)idx * 27u;
        float s = 0.f;
#pragma unroll
        for (int t = 0; t < 27; ++t) s += p[t];
        sw[idx] = s;
    }
}

// ---------------------------------------------------------------------------
// Phase 3: D[b,o] = sum_c Sx[b,c] * Sw[o,c]   (M=16, N=64, K=32)
// via V_WMMA_F32_16X16X4_F32, 8 K-steps. One wave32 per 16-wide N tile.
// Fused epilogue: ((D/34^3 + bias)*2 - mean) * rsqrt(var + eps)
// ---------------------------------------------------------------------------
__global__ __launch_bounds__(128) void gemm_bn_kernel(const float* __restrict__ sx,
                                                      const float* __restrict__ sw,
                                                      const float* __restrict__ bias,
                                                      const float* __restrict__ rmean,
                                                      const float* __restrict__ rvar,
                                                      float* __restrict__ out) {
    const int lane  = threadIdx.x & 31;
    const int wave  = threadIdx.x >> 5;       // N tile: 0..3
    const int m     = lane & 15;              // A row / D col index base
    const int khalf = lane >> 4;              // 0: K={0,1}, 1: K={2,3} within step
    const int n     = (wave << 4) + (lane & 15);  // global output channel

    v8f acc = {};
#pragma unroll
    for (int k0 = 0; k0 < 32; k0 += 4) {
        const int ks = k0 + 2 * khalf;
        v2f a, b;
        a.x = sx[m * 32 + ks];
        a.y = sx[m * 32 + ks + 1];
        b.x = sw[n * 32 + ks];
        b.y = sw[n * 32 + ks + 1];
        acc = __builtin_amdgcn_wmma_f32_16x16x4_f32(
            /*neg_a=*/false, a, /*neg_b=*/false, b,
            /*c_mod=*/(short)0, acc, /*reuse_a=*/false, /*reuse_b=*/false);
    }

    const float invVol = 1.0f / (34.0f * 34.0f * 34.0f);
    const float bo = bias[n];
    const float mo = rmean[n];
    const float is = rsqrtf(rvar[n] + 1e-5f);
#pragma unroll
    for (int v = 0; v < 8; ++v) {
        const int row = v + 8 * khalf;  // batch index b
        const float val = ((acc[v] * invVol + bo) * 2.0f - mo) * is;
        out[row * 64 + n] = val;
    }
}

extern "C" void kernel_launch(void* const* d_in, const int* in_sizes, int n_in,
                              void* d_out, int out_size, void* d_ws, size_t ws_size,
                              hipStream_t stream) {
    (void)in_sizes; (void)n_in; (void)out_size; (void)ws_size;
    const float* x     = (const float*)d_in[0];  // (16,32,32,32,32)
    const float* w     = (const float*)d_in[1];  // (64,32,3,3,3)
    const float* bias  = (const float*)d_in[2];  // (64,)
    const float* rmean = (const float*)d_in[3];  // (64,)
    const float* rvar  = (const float*)d_in[4];  // (64,)
    float* out = (float*)d_out;                  // (16,64)

    float* sx = (float*)d_ws;                    // 512 floats
    float* sw = (float*)((char*)d_ws + 4096);    // 2048 floats

    reduce_x_kernel<<<512, 256, 0, stream>>>(x, sx);
    reduce_w_kernel<<<8, 256, 0, stream>>>(w, sw);
    gemm_bn_kernel<<<1, 128, 0, stream>>>(sx, sw, bias, rmean, rvar, out);
}